// AddSparseAndLowRankCorrectionFP32_83159156785479
// MI455X (gfx1250) — compile-verified
//
#include <hip/hip_runtime.h>
#include <hip/hip_fp16.h>

// ---------------------------------------------------------------------------
// AddSparseAndLowRankCorrection, MI455X (gfx1250, wave32, WMMA)
//
// out = x @ W_eff^T + b,  W_eff = W_inner + fp16rt(A) fp16rt(B) + S_dense
// Single merged fp32 GEMM (275 GFLOP) on V_WMMA_F32_16X16X4_F32.
// ---------------------------------------------------------------------------

#define D_IN   4096
#define D_OUT  4096
#define RANK   64

typedef __attribute__((ext_vector_type(2))) float v2f;
typedef __attribute__((ext_vector_type(8))) float v8f;

__device__ __forceinline__ float fp16rt(float x) {
    return __half2float(__float2half(x));
}

__device__ __forceinline__ v8f wmma4(v2f a, v2f b, v8f c) {
    // 8 args: (neg_a, A, neg_b, B, c_mod, C, reuse_a, reuse_b)
    return __builtin_amdgcn_wmma_f32_16x16x4_f32(
        false, a, false, b, (short)0, c, false, false);
}

// ---------------------------------------------------------------------------
// Kernel 1: W_eff[o,d] = W_inner[o,d] + sum_r fp16rt(A[o,r]) * fp16rt(B[r,d])
// A: [D_OUT, RANK], B: [RANK, D_IN], both row-major and cache-resident (1 MB).
// ---------------------------------------------------------------------------
__global__ __launch_bounds__(256) void build_weff_kernel(
    const float* __restrict__ Wi, const float* __restrict__ A,
    const float* __restrict__ Bm, float* __restrict__ Weff)
{
    unsigned idx = blockIdx.x * 256u + threadIdx.x;   // 0 .. 16M-1
    unsigned o = idx >> 12;          // row    (D_IN == 4096)
    unsigned d = idx & 4095u;        // column

    float acc = 0.0f;
    const float* Arow = A + (size_t)o * RANK;
#pragma unroll 8
    for (int r = 0; r < RANK; ++r) {
        acc += fp16rt(Arow[r]) * fp16rt(Bm[(size_t)r * D_IN + d]);
    }
    Weff[idx] = Wi[idx] + acc;
}

// ---------------------------------------------------------------------------
// Kernel 2: scatter-add the COO correction (duplicates must sum -> atomics).
// sparse_indices is int64: rows = idx[0:nnz], cols = idx[nnz:2*nnz].
// ---------------------------------------------------------------------------
__global__ __launch_bounds__(256) void scatter_coo_kernel(
    const float* __restrict__ vals, const long long* __restrict__ idx,
    float* __restrict__ Weff, int nnz)
{
    int i = blockIdx.x * 256 + threadIdx.x;
    if (i >= nnz) return;
    long long r = idx[i];
    long long c = idx[(size_t)nnz + i];
    atomicAdd(Weff + (size_t)r * D_IN + c, fp16rt(vals[i]));
}

// ---------------------------------------------------------------------------
// Kernel 3: out[m, n] = sum_k x[m, k] * Weff[n, k] + bias[n]
//
// Wave32 tiling: one wave owns a 32(M) x 64(N) tile = 2 A-fragments x
// 4 B-fragments = 8 independent accumulators. Per K-step of 4:
// 6 float2 loads feed 8 V_WMMA_F32_16X16X4_F32 (0.75 loads/WMMA), and no
// two consecutive WMMAs share a D register, keeping the XDL pipe busy.
//
// ISA 16x16x4 F32 layouts (CDNA5 ISA):
//   A (16x4):  lanes 0-15 -> M=lane,  v0=K0 v1=K1; lanes 16-31 -> v0=K2 v1=K3
//   B (4x16):  lanes 0-15 -> N=lane,  v0=K0 v1=K1; lanes 16-31 -> v0=K2 v1=K3
//   C (16x16): VGPR v: lanes 0-15 -> M=v, N=lane; lanes 16-31 -> M=v+8
// Both fragments are contiguous per-lane float2 loads (row-major Weff == B^T).
// ---------------------------------------------------------------------------
__global__ __launch_bounds__(256) void gemm_wmma_kernel(
    const float* __restrict__ X, const float* __restrict__ Weff,
    const float* __restrict__ bias, float* __restrict__ out)
{
    const int lane = threadIdx.x & 31;
    const int wave = threadIdx.x >> 5;                 // 0..7
    const int wg   = blockIdx.x * 8 + wave;            // global wave id
    const int m0   = (wg >> 6) * 32;                   // 64 n-groups per m-tile
    const int n0   = (wg & 63) * 64;

    const int lm   = lane & 15;
    const int half = lane >> 4;                        // 0 or 1 (K 0..1 vs 2..3)

    const float* xp0 = X + (size_t)(m0 + lm) * D_IN + 2 * half;
    const float* xp1 = xp0 + (size_t)16 * D_IN;
    const float* wp  = Weff + (size_t)(n0 + lm) * D_IN + 2 * half;

    v8f acc[2][4];
#pragma unroll
    for (int mi = 0; mi < 2; ++mi)
#pragma unroll
        for (int t = 0; t < 4; ++t) acc[mi][t] = (v8f){};

    for (int k = 0; k < D_IN; k += 16) {
        // keep the streamed x rows ahead in near cache (Weff lives in L2)
        __builtin_prefetch(xp0 + k + 512, 0, 3);
        __builtin_prefetch(xp1 + k + 512, 0, 3);
#pragma unroll
        for (int kk = 0; kk < 16; kk += 4) {
            v2f a0 = *(const v2f*)(xp0 + k + kk);
            v2f a1 = *(const v2f*)(xp1 + k + kk);
            v2f b0 = *(const v2f*)(wp + k + kk);
            v2f b1 = *(const v2f*)(wp + 16 * D_IN + k + kk);
            v2f b2 = *(const v2f*)(wp + 32 * D_IN + k + kk);
            v2f b3 = *(const v2f*)(wp + 48 * D_IN + k + kk);

            acc[0][0] = wmma4(a0, b0, acc[0][0]);
            acc[1][0] = wmma4(a1, b0, acc[1][0]);
            acc[0][1] = wmma4(a0, b1, acc[0][1]);
            acc[1][1] = wmma4(a1, b1, acc[1][1]);
            acc[0][2] = wmma4(a0, b2, acc[0][2]);
            acc[1][2] = wmma4(a1, b2, acc[1][2]);
            acc[0][3] = wmma4(a0, b3, acc[0][3]);
            acc[1][3] = wmma4(a1, b3, acc[1][3]);
        }
    }

    // Epilogue: + bias[n], scatter C layout back to row-major out.
#pragma unroll
    for (int mi = 0; mi < 2; ++mi) {
#pragma unroll
        for (int t = 0; t < 4; ++t) {
            const int n = n0 + t * 16 + lm;
            const float bv = bias[n];
            float* op = out + (size_t)(m0 + mi * 16 + half * 8) * D_OUT + n;
#pragma unroll
            for (int v = 0; v < 8; ++v) {
                op[(size_t)v * D_OUT] = acc[mi][t][v] + bv;
            }
        }
    }
}

// ---------------------------------------------------------------------------
// Inputs (setup_inputs order):
//   0: x              [4, 2048, 4096]  f32
//   1: W_inner        [4096, 4096]     f32
//   2: b_inner        [4096]           f32
//   3: A              [4096, 64]       f32
//   4: B              [64, 4096]       f32
//   5: sparse_values  [NNZ]            f32
//   6: sparse_indices [2, NNZ]         i64
// Output: [4, 2048, 4096] f32.   d_ws holds W_eff (64 MB).
// ---------------------------------------------------------------------------
extern "C" void kernel_launch(void* const* d_in, const int* in_sizes, int n_in,
                              void* d_out, int out_size, void* d_ws, size_t ws_size,
                              hipStream_t stream) {
    const float*     x    = (const float*)d_in[0];
    const float*     Wi   = (const float*)d_in[1];
    const float*     b    = (const float*)d_in[2];
    const float*     A    = (const float*)d_in[3];
    const float*     Bm   = (const float*)d_in[4];
    const float*     sv   = (const float*)d_in[5];
    const long long* sidx = (const long long*)d_in[6];
    float*           out  = (float*)d_out;
    float*           Weff = (float*)d_ws;

    const int nnz = in_sizes[5];
    const int M   = in_sizes[0] / D_IN;       // 8192 tokens

    // 1) W_eff = W_inner + fp16rt(A) @ fp16rt(B)
    build_weff_kernel<<<(D_OUT * D_IN) / 256, 256, 0, stream>>>(Wi, A, Bm, Weff);

    // 2) W_eff += densify(sparse)
    scatter_coo_kernel<<<(nnz + 255) / 256, 256, 0, stream>>>(sv, sidx, Weff, nnz);

    // 3) out = x @ W_eff^T + b   (fp32 WMMA, 32x64 tile per wave)
    const int m_tiles     = M / 32;            // 256
    const int n_groups    = D_OUT / 64;        // 64
    const int total_waves = m_tiles * n_groups;
    gemm_wmma_kernel<<<total_waves / 8, 256, 0, stream>>>(x, Weff, b, out);
}